// symmetry_module_27728308863508
// MI455X (gfx1250) — compile-verified
//
#include <hip/hip_runtime.h>
#include <hip/hip_bf16.h>
#include <math.h>

// MI455X (gfx1250) — wave32, WMMA f32 16x16x4.
// One block per n (256 blocks, 128 threads = 4 waves).
// Each wave owns one 16x16 quadrant of the (padded) 32x32 Gram matrix for
// all 10 spherical-harmonic channels. Phase tiles (64 x 32, f32) are staged
// in LDS transposed (Pt[v][t]) so WMMA fragment fetches are contiguous b64
// LDS loads. Trig intermediates are computed once per element and shared
// across the 10 channels (transcendental cost dominates, not FLOPs).

typedef __attribute__((ext_vector_type(2))) float v2f;
typedef __attribute__((ext_vector_type(8))) float v8f;

#define N_   256
#define T_   1024
#define V_   25
#define NCH  10
#define KC   64          // T-chunk rows staged per iteration
#define PTS  68          // padded t-stride for Pt[32][KC]
#define NTHR 128

__global__ __launch_bounds__(NTHR)
void sph_gram_kernel(const float* __restrict__ x, float* __restrict__ out) {
    __shared__ float lds_sth[KC * V_];
    __shared__ float lds_cth[KC * V_];
    __shared__ float lds_c  [KC * V_];
    __shared__ float lds_s  [KC * V_];
    __shared__ float Pt[32 * PTS];        // phase tile, transposed, padded

    const int n    = blockIdx.x;
    const int tid  = threadIdx.x;
    const int lane = tid & 31;
    const int wv   = tid >> 5;            // wave id 0..3
    const int ti   = wv >> 1;             // row quadrant
    const int tj   = wv & 1;              // col quadrant

    // Zero the pad rows v=25..31 once; never rewritten afterwards.
    for (int i = tid; i < 7 * KC; i += NTHR) {
        int v = 25 + (i >> 6);
        int t = i & (KC - 1);
        Pt[v * PTS + t] = 0.0f;
    }

    v8f acc[NCH];
#pragma unroll
    for (int ch = 0; ch < NCH; ++ch) acc[ch] = (v8f){};

    // Fragment base addresses (fixed per lane across whole kernel).
    const int rowA = ti * 16 + (lane & 15);
    const int rowB = tj * 16 + (lane & 15);
    const int koff = (lane >> 4) * 2;     // lanes 16..31 hold K=2,3
    const float* pA = &Pt[rowA * PTS + koff];
    const float* pB = &Pt[rowB * PTS + koff];

    const size_t xbase = (size_t)n * 3 * T_ * V_;

    for (int t0 = 0; t0 < T_; t0 += KC) {
        __syncthreads();   // protect intermediates (WAR vs previous chunk)

        // ---- Stage 1: per-element trig intermediates, shared by all 10 ch.
        for (int e = tid; e < KC * V_; e += NTHR) {
            int t = e & (KC - 1);
            int v = e >> 6;
            int gt = t0 + t;
            float x0 = x[xbase + 0 * (size_t)T_ * V_ + (size_t)gt * V_ + v];
            float x1 = x[xbase + 1 * (size_t)T_ * V_ + (size_t)gt * V_ + v];
            float x2 = x[xbase + 2 * (size_t)T_ * V_ + (size_t)gt * V_ + v];
            float xy    = x0 * x0 + x1 * x1;
            float theta = atan2f(sqrtf(xy) + 1e-5f, x2 + 1e-5f);
            float phi   = atan2f(x1 + 1e-5f, x0 + 1e-5f);
            float sth, cth;
            sincosf(theta, &sth, &cth);
            float c = cosf(phi);
            float s = sqrtf(fminf(fmaxf(1.0f - c * c, 1e-12f), 1.0f));
            lds_sth[e] = sth;
            lds_cth[e] = cth;
            lds_c[e]   = c;
            lds_s[e]   = s;
        }
        __syncthreads();

        // ---- Stage 2: per channel, build phase tile then WMMA-accumulate.
#pragma unroll
        for (int ch = 0; ch < NCH; ++ch) {
            for (int e = tid; e < KC * V_; e += NTHR) {
                float sth = lds_sth[e];
                float cth = lds_cth[e];
                float c   = lds_c[e];
                float s   = lds_s[e];
                float amp, sm, cm;
                switch (ch) {   // (l,m): norm * P_l^m(c), angle multiple m
                case 0: amp =  0.28209479f;                          sm = 0.0f; cm = 1.0f; break; // (0,0)
                case 1: amp =  0.48860251f * c;                      sm = 0.0f; cm = 1.0f; break; // (1,0)
                case 2: amp = -0.34549415f * s;                      sm = sth;  cm = cth;  break; // (1,1)
                case 3: amp =  0.63078313f * (1.5f * c * c - 0.5f);  sm = 0.0f; cm = 1.0f; break; // (2,0)
                case 4: amp = -0.77254840f * c * s;                  sm = sth;  cm = cth;  break; // (2,1)
                case 5: amp =  0.38627420f * (1.0f - c * c);
                        sm = 2.0f * sth * cth; cm = 1.0f - 2.0f * sth * sth;                break; // (2,2)
                case 6: amp =  0.74635267f * c * (2.5f * c * c - 1.5f); sm = 0.0f; cm = 1.0f; break; // (3,0)
                case 7: amp = -0.32318014f * (5.0f * c * c - 1.0f) * s; sm = sth;  cm = cth;  break; // (3,1)
                case 8: amp =  1.02198547f * c * (1.0f - c * c);
                        sm = 2.0f * sth * cth; cm = 1.0f - 2.0f * sth * sth;                break; // (3,2)
                default: amp = -0.41722381f * s * s * s;
                        sm = sth * (3.0f - 4.0f * sth * sth);
                        cm = cth * (4.0f * cth * cth - 3.0f);                               break; // (3,3)
                }
                float ph = atan2f(amp * sm, amp * cm);
                int t = e & (KC - 1);
                int v = e >> 6;
                Pt[v * PTS + t] = ph;
            }
            __syncthreads();

            // 16 WMMAs over the 64-deep K chunk (K=4 per instruction).
#pragma unroll
            for (int k0 = 0; k0 < KC; k0 += 4) {
                v2f a, b;
                a.x = pA[k0]; a.y = pA[k0 + 1];
                b.x = pB[k0]; b.y = pB[k0 + 1];
                acc[ch] = __builtin_amdgcn_wmma_f32_16x16x4_f32(
                    false, a, false, b, (short)0, acc[ch], false, false);
            }
            __syncthreads();   // all waves done reading Pt before next fill
        }
    }

    // ---- Writeback: C/D layout — VGPR r, lanes 0-15 => M=r, lanes 16-31 => M=r+8.
    const int wcol  = tj * 16 + (lane & 15);
    const int mbase = ti * 16 + 8 * (lane >> 4);
    if (wcol < V_) {
#pragma unroll
        for (int ch = 0; ch < NCH; ++ch) {
#pragma unroll
            for (int r = 0; r < 8; ++r) {
                int vrow = mbase + r;
                if (vrow < V_) {
                    out[(((size_t)n * NCH + ch) * V_ + vrow) * V_ + wcol] = acc[ch][r];
                }
            }
        }
    }
}

extern "C" void kernel_launch(void* const* d_in, const int* in_sizes, int n_in,
                              void* d_out, int out_size, void* d_ws, size_t ws_size,
                              hipStream_t stream) {
    (void)in_sizes; (void)n_in; (void)out_size; (void)d_ws; (void)ws_size;
    const float* x = (const float*)d_in[0];
    float* out = (float*)d_out;
    sph_gram_kernel<<<N_, NTHR, 0, stream>>>(x, out);
}